// QuantumConvLayer_45818711114236
// MI455X (gfx1250) — compile-verified
//
#include <hip/hip_runtime.h>
#include <hip/hip_bf16.h>
#include <stdint.h>

// Memory-bound elementwise kernel (quantum 2-qubit probs).
// CDNA5 path: async global->LDS staging (ASYNCcnt) with double buffering,
// NT temporal hints both directions, b128 vector access.

typedef float f32x4 __attribute__((ext_vector_type(4)));

#define BLOCK 256
#define ITERS 8

__global__ __launch_bounds__(BLOCK) void quantum_probs_async_kernel(
    const f32x4* __restrict__ in, f32x4* __restrict__ out, int nrows)
{
    __shared__ f32x4 stage[2][BLOCK];

    const int tid = threadIdx.x;
    const long stride = (long)gridDim.x * BLOCK;          // rows per grid-tile
    long row = (long)blockIdx.x * BLOCK + tid;            // my row in tile 0
    long tile_base = (long)blockIdx.x * BLOCK;            // uniform per block

    // Low 32 bits of a generic pointer to __shared__ == LDS byte offset
    // (hardware adds the wave's LDS_BASE).
    const uint32_t lds0 = (uint32_t)(uintptr_t)(void*)&stage[0][tid];
    const uint32_t lds1 = (uint32_t)(uintptr_t)(void*)&stage[1][tid];

    // ---- Prologue: async-load tile 0 into buffer 0 (ASYNCcnt++) ----
    if (row < nrows) {
        uint64_t g = (uint64_t)(uintptr_t)(in + row);
        asm volatile("global_load_async_to_lds_b128 %0, %1, off th:TH_LOAD_NT"
                     :: "v"(lds0), "v"(g) : "memory");
    }

    int buf = 0;
#pragma unroll
    for (int it = 0; it < ITERS; ++it) {
        const long next_tile = tile_base + stride;

        if (next_tile < nrows) {                          // uniform branch
            // Prefetch tile it+1 into the other buffer, then wait for tile it.
            const long next = row + stride;
            if (next < nrows) {                           // per-lane EXEC mask
                uint64_t g = (uint64_t)(uintptr_t)(in + next);
                uint32_t l = buf ? lds0 : lds1;
                asm volatile("global_load_async_to_lds_b128 %0, %1, off th:TH_LOAD_NT"
                             :: "v"(l), "v"(g) : "memory");
            }
            // Async loads complete in order: <=1 outstanding => tile `it` arrived.
            asm volatile("s_wait_asynccnt 1" ::: "memory");
        } else {
            // Last tile for this block: drain everything.
            asm volatile("s_wait_asynccnt 0" ::: "memory");
        }

        if (row < nrows) {
            const f32x4 v = stage[buf][tid];              // ds_load_b128
            const float theta = v.x;
            // cos^2(t/2) = 0.5 + 0.5*cos(t), sin^2(t/2) = 0.5 - 0.5*cos(t)
            const float c  = __cosf(theta);               // v_cos_f32 (|theta|~N(0,1))
            const float c2 = fmaf(0.5f, c, 0.5f);
            const float s2 = fmaf(-0.5f, c, 0.5f);
            f32x4 o;
            o.x = c2 * c2;                                // P(00)
            o.y = c2 * s2;                                // P(01)
            o.z = s2 * s2;                                // P(11)
            o.w = s2 * c2;                                // P(10)
            __builtin_nontemporal_store(o, out + row);    // global_store_b128 NT
        }

        row += stride;
        tile_base += stride;
        buf ^= 1;
    }
}

extern "C" void kernel_launch(void* const* d_in, const int* in_sizes, int n_in,
                              void* d_out, int out_size, void* d_ws, size_t ws_size,
                              hipStream_t stream) {
    (void)n_in; (void)d_ws; (void)ws_size; (void)out_size;
    const f32x4* in  = (const f32x4*)d_in[0];
    f32x4*       out = (f32x4*)d_out;
    const int nrows = in_sizes[0] / 4;                    // [N,4] flattened

    const int rows_per_block = BLOCK * ITERS;             // 2048 rows/block
    const int blocks = (nrows + rows_per_block - 1) / rows_per_block;  // 8192 for N=2^24

    quantum_probs_async_kernel<<<blocks, BLOCK, 0, stream>>>(in, out, nrows);
}